// CGBN_cuda2_3813930959611
// MI455X (gfx1250) — compile-verified
//
#include <hip/hip_runtime.h>
#include <math.h>

// ---------------- static structure (mirrors the reference) ----------------
#define NTRI 42
#define BATCH 256

// triples sorted by (l, l1, l2)
__constant__ int TRI_L [NTRI] = {0,0,0,0,0, 1,1,1,1,1,1,1,1, 2,2,2,2,2,2,2,2,2,2, 3,3,3,3,3,3,3,3,3,3, 4,4,4,4,4,4,4,4,4};
__constant__ int TRI_L1[NTRI] = {0,1,2,3,4, 1,1,2,2,3,3,4,4, 1,2,2,2,3,3,3,4,4,4, 2,2,3,3,3,3,4,4,4,4, 2,3,3,3,4,4,4,4,4};
__constant__ int TRI_L2[NTRI] = {0,1,2,3,4, 0,1,1,2,2,3,3,4, 1,0,1,2,1,2,3,2,3,4, 1,2,0,1,2,3,1,2,3,4, 2,1,2,3,0,1,2,3,4};
// channel base within the degree (256 channels per triple, in sorted order)
__constant__ int TRI_CHB[NTRI] = {0,256,512,768,1024,
                                  0,256,512,768,1024,1280,1536,1792,
                                  0,256,512,768,1024,1280,1536,1792,2048,2304,
                                  0,256,512,768,1024,1280,1536,1792,2048,2304,
                                  0,256,512,768,1024,1280,1536,1792,2048};

__constant__ int TFF [5] = {1280,2048,2560,2560,2304};          // K per degree
__constant__ int DEGC[6] = {0,1280,3328,5888,8448,10752};       // channel offset per degree
__constant__ int FOFF[5] = {0,16,64,144,256};                   // per-degree offset in 400-dim (in & out)
__constant__ long long XOFS[5] = {0,327680,1900544,5177344,9764864}; // per-degree offset in X planes (floats)
__constant__ int WOFS[5] = {0,20480,53248,94208,135168};        // per-degree offset in W planes (=16*DEGC)

// compile-time mirrors for the templated GEMM
constexpr int       cTFF [5] = {1280,2048,2560,2560,2304};
constexpr int       cFOFF[5] = {0,16,64,144,256};
constexpr long long cXOFS[5] = {0,327680,1900544,5177344,9764864};
constexpr int       cWOFS[5] = {0,20480,53248,94208,135168};

#define PLANE 15073280ll        // floats per X plane (real or imag)
#define NCH   10752             // total fragment channels
#define WPLN  172032            // floats per scaled-weight plane

typedef float v2f __attribute__((ext_vector_type(2)));
typedef float v8f __attribute__((ext_vector_type(8)));

// ---------------- Clebsch-Gordan coefficients (device, double precision) ----------------
__device__ double dfact(int n) { double r = 1.0; for (int i = 2; i <= n; ++i) r *= (double)i; return r; }

__device__ float cg_coef(int j1, int m1, int j2, int m2, int j, int m) {
    if (m1 + m2 != m) return 0.f;
    if (j < (j1 > j2 ? j1 - j2 : j2 - j1) || j > j1 + j2) return 0.f;
    double pref = sqrt((2.0 * j + 1.0) * dfact(j + j1 - j2) * dfact(j - j1 + j2) *
                       dfact(j1 + j2 - j) / dfact(j1 + j2 + j + 1));
    pref *= sqrt(dfact(j + m) * dfact(j - m) * dfact(j1 - m1) * dfact(j1 + m1) *
                 dfact(j2 - m2) * dfact(j2 + m2));
    double s = 0.0;
    for (int k = 0; k <= j1 + j2 - j; ++k) {
        if (j1 - m1 - k < 0 || j2 + m2 - k < 0 || j - j2 + m1 + k < 0 || j - j1 - m2 + k < 0) continue;
        double d = dfact(k) * dfact(j1 + j2 - j - k) * dfact(j1 - m1 - k) *
                   dfact(j2 + m2 - k) * dfact(j - j2 + m1 + k) * dfact(j - j1 - m2 + k);
        s += ((k & 1) ? -1.0 : 1.0) / d;
    }
    return (float)(pref * s);
}

// one thread per triple: build sparse (m,p,q,coef) list, fixed enumeration order -> deterministic
__global__ void k_cg_init(unsigned* __restrict__ cg_pack, float* __restrict__ cg_val, int* __restrict__ cg_cnt) {
    int t = blockIdx.x * blockDim.x + threadIdx.x;
    if (t >= NTRI) return;
    int l = TRI_L[t], l1 = TRI_L1[t], l2 = TRI_L2[t];
    int cnt = 0;
    for (int m = -l; m <= l; ++m)
        for (int m1 = -l1; m1 <= l1; ++m1) {
            int m2 = m - m1;
            if (m2 < -l2 || m2 > l2) continue;
            float c = cg_coef(l1, m1, l2, m2, l, m);
            if (c == 0.f) continue;
            cg_pack[t * 96 + cnt] = (unsigned)((m + l) | ((m1 + l1) << 8) | ((m2 + l2) << 16));
            cg_val [t * 96 + cnt] = c;
            ++cnt;
        }
    cg_cnt[t] = cnt;
}

// ---------------- pass 1: fragments -> X planes [(b,m) x c] + per-(c,b) |z|^2 partials ----------------
__global__ __launch_bounds__(256) void k_frag(const float* __restrict__ Fs,
                                              float* __restrict__ Xr, float* __restrict__ Xi,
                                              float* __restrict__ vpart,
                                              const unsigned* __restrict__ cg_pack,
                                              const float* __restrict__ cg_val,
                                              const int* __restrict__ cg_cnt) {
    __shared__ float sA[16 * 9 * 2];
    __shared__ float sB[16 * 9 * 2];
    __shared__ float sFrag[256 * 18];
    __shared__ unsigned sPk[96];
    __shared__ float sVl[96];
    __shared__ int sCnt;

    int t = blockIdx.x % NTRI;
    int b = blockIdx.x / NTRI;
    int tid = threadIdx.x;
    int l = TRI_L[t], l1 = TRI_L1[t], l2 = TRI_L2[t];
    int na = 2 * l1 + 1, nb = 2 * l2 + 1, nm = 2 * l + 1;

    const float* pa = Fs + ((size_t)b * 400 + FOFF[l1]) * 2;
    const float* pb = Fs + ((size_t)b * 400 + FOFF[l2]) * 2;
    int nAf = 16 * na * 2, nBf = 16 * nb * 2;
    for (int idx = tid; idx < nAf; idx += 256) sA[idx] = pa[idx];
    for (int idx = tid; idx < nBf; idx += 256) sB[idx] = pb[idx];
    if (tid == 0) sCnt = cg_cnt[t];
    for (int idx = tid; idx < 96; idx += 256) { sPk[idx] = cg_pack[t * 96 + idx]; sVl[idx] = cg_val[t * 96 + idx]; }
    float* fr = &sFrag[tid * 18];
    for (int m = 0; m < nm; ++m) { fr[m * 2] = 0.f; fr[m * 2 + 1] = 0.f; }
    __syncthreads();

    int i = tid >> 4, j = tid & 15;
    int cnt = sCnt;
    for (int e = 0; e < cnt; ++e) {
        unsigned pk = sPk[e];
        float c = sVl[e];
        int m = pk & 0xff, p = (pk >> 8) & 0xff, q = (pk >> 16) & 0xff;
        float ar = sA[(i * na + p) * 2], ai = sA[(i * na + p) * 2 + 1];
        float br = sB[(j * nb + q) * 2], bi = sB[(j * nb + q) * 2 + 1];
        fr[m * 2]     += c * (ar * br - ai * bi);
        fr[m * 2 + 1] += c * (ar * bi + ai * br);
    }

    int K = TFF[l];
    int ch = TRI_CHB[t] + tid;
    long long xo = XOFS[l];
    float vs = 0.f;
    for (int m = 0; m < nm; ++m) {
        float re = fr[m * 2], im = fr[m * 2 + 1];
        long long row = (long long)(b * nm + m);
        Xr[xo + row * K + ch] = re;
        Xi[xo + row * K + ch] = im;
        vs += re * re + im * im;
    }
    vpart[(size_t)(DEGC[l] + ch) * BATCH + b] = vs;   // unique writer -> deterministic reduction later
}

// ---------------- pass 1.5: reduce variance (fixed order) + fold 1/(sqrt(var)+eps) into weights ----------------
// scaled weights stored in WMMA-B-swizzled layout: idx = (c>>1)*32 + o*2 + (c&1)
__global__ __launch_bounds__(256) void k_scale(const float* __restrict__ W, const float* __restrict__ vpart,
                                               float* __restrict__ Wr, float* __restrict__ Wi,
                                               float* __restrict__ Wn) {
    int c = blockIdx.x * 256 + threadIdx.x;
    if (c >= NCH) return;
    int l = 0;
    while (l < 4 && c >= DEGC[l + 1]) ++l;
    int cd = c - DEGC[l];
    int nm = 2 * l + 1, K = TFF[l];
    const float* vp = vpart + (size_t)c * BATCH;
    float s = 0.f;
    for (int b = 0; b < BATCH; ++b) s += vp[b];
    float mean = s / (256.f * (float)nm);
    float inv = 1.f / (sqrtf(mean) + 1e-5f);
    for (int o = 0; o < 16; ++o) {
        size_t win = ((size_t)WOFS[l] + (size_t)o * K + cd) * 2;
        float wr = W[win] * inv, wim = W[win + 1] * inv;
        size_t idx = (size_t)WOFS[l] + (size_t)(cd >> 1) * 32 + o * 2 + (cd & 1);
        Wr[idx] = wr; Wi[idx] = wim; Wn[idx] = -wim;
    }
}

// ---------------- pass 2: complex GEMM via V_WMMA_F32_16X16X4_F32 ----------------
// templated on degree L: K, nm, offsets are compile-time -> uniform scalar loop control,
// strength-reduced epilogue, unrollable K loop. One wave per 16x16 output tile.
template <int L>
__global__ __launch_bounds__(256) void k_gemm(const float* __restrict__ Xr, const float* __restrict__ Xi,
                                              const float* __restrict__ Wr, const float* __restrict__ Wi,
                                              const float* __restrict__ Wn, float* __restrict__ out) {
    constexpr int       K    = cTFF[L];
    constexpr int       NM   = 2 * L + 1;
    constexpr long long XO   = cXOFS[L];
    constexpr int       WO   = cWOFS[L];
    constexpr int       OOFS = cFOFF[L];

    int tile = blockIdx.x * 8 + ((int)threadIdx.x >> 5);   // grid sized so all waves are valid
    int lane = threadIdx.x & 31;
    int n    = lane & 15;      // A row / B col / D col
    int half = lane >> 4;      // which K pair of the 4-wide step

    const float* ar_p = Xr + XO + (size_t)(tile * 16 + n) * K + half * 2;
    const float* ai_p = Xi + XO + (size_t)(tile * 16 + n) * K + half * 2;
    const float* br_p = Wr + WO + (size_t)half * 32 + n * 2;
    const float* bi_p = Wi + WO + (size_t)half * 32 + n * 2;
    const float* bn_p = Wn + WO + (size_t)half * 32 + n * 2;

    v8f accR = {};
    v8f accI = {};
    for (int k0 = 0; k0 < K; k0 += 16) {
        // keep the streaming X planes ~4KB ahead of the WMMA pipe (global_prefetch_b8)
        __builtin_prefetch(ar_p + k0 + 1024, 0, 1);
        __builtin_prefetch(ai_p + k0 + 1024, 0, 1);
#pragma unroll
        for (int kk = 0; kk < 16; kk += 4) {
            int k = k0 + kk;
            v2f a_r = *(const v2f*)(ar_p + k);
            v2f a_i = *(const v2f*)(ai_p + k);
            v2f b_r = *(const v2f*)(br_p + k * 16);   // (k>>1)*32
            v2f b_i = *(const v2f*)(bi_p + k * 16);
            v2f b_n = *(const v2f*)(bn_p + k * 16);
            // yr = Xr*Wr' - Xi*Wi'   (f32 WMMA has no A/B neg -> use pre-negated Wi')
            accR = __builtin_amdgcn_wmma_f32_16x16x4_f32(false, a_r, false, b_r, (short)0, accR, false, false);
            accR = __builtin_amdgcn_wmma_f32_16x16x4_f32(false, a_i, false, b_n, (short)0, accR, false, false);
            // yi = Xr*Wi' + Xi*Wr'
            accI = __builtin_amdgcn_wmma_f32_16x16x4_f32(false, a_r, false, b_i, (short)0, accI, false, false);
            accI = __builtin_amdgcn_wmma_f32_16x16x4_f32(false, a_i, false, b_r, (short)0, accI, false, false);
        }
    }

#pragma unroll
    for (int v = 0; v < 8; ++v) {
        int r = tile * 16 + half * 8 + v;        // D row for this VGPR
        int bb = r / NM, m = r % NM;             // constant divisor -> mul/shift
        size_t o = (((size_t)bb * 400) + OOFS + n * NM + m) * 2;
        out[o]     = accR[v];
        out[o + 1] = accI[v];
    }
}

// ---------------- host side ----------------
extern "C" void kernel_launch(void* const* d_in, const int* in_sizes, int n_in,
                              void* d_out, int out_size, void* d_ws, size_t ws_size,
                              hipStream_t stream) {
    (void)in_sizes; (void)n_in; (void)out_size;
    const float* Fs = (const float*)d_in[0];   // [256, 400, 2] f32
    const float* W  = (const float*)d_in[1];   // [172032, 2]   f32
    float* out = (float*)d_out;                // [256, 400, 2] f32
    float* ws = (float*)d_ws;

    // workspace layout (floats)
    float*    Xr      = ws;                          // PLANE
    float*    Xi      = Xr + PLANE;                  // PLANE
    float*    vpart   = Xi + PLANE;                  // NCH*BATCH = 2,752,512
    float*    Wrp     = vpart + (size_t)NCH * BATCH; // WPLN
    float*    Wip     = Wrp + WPLN;                  // WPLN
    float*    Wnp     = Wip + WPLN;                  // WPLN
    float*    cg_val  = Wnp + WPLN;                  // 42*96
    unsigned* cg_pack = (unsigned*)(cg_val + NTRI * 96);
    int*      cg_cnt  = (int*)(cg_pack + NTRI * 96);
    size_t need = (size_t)(cg_cnt + NTRI - (int*)ws) * sizeof(float);
    if (ws_size < need) return;   // workspace too small; cannot run

    k_cg_init<<<1, 64, 0, stream>>>(cg_pack, cg_val, cg_cnt);
    k_frag  <<<BATCH * NTRI, 256, 0, stream>>>(Fs, Xr, Xi, vpart, cg_pack, cg_val, cg_cnt);
    k_scale <<<(NCH + 255) / 256, 256, 0, stream>>>(W, vpart, Wrp, Wip, Wnp);
    // tiles per degree: 16 / 48 / 80 / 112 / 144 ; 8 waves (tiles) per block
    k_gemm<0><<< 2, 256, 0, stream>>>(Xr, Xi, Wrp, Wip, Wnp, out);
    k_gemm<1><<< 6, 256, 0, stream>>>(Xr, Xi, Wrp, Wip, Wnp, out);
    k_gemm<2><<<10, 256, 0, stream>>>(Xr, Xi, Wrp, Wip, Wnp, out);
    k_gemm<3><<<14, 256, 0, stream>>>(Xr, Xi, Wrp, Wip, Wnp, out);
    k_gemm<4><<<18, 256, 0, stream>>>(Xr, Xi, Wrp, Wip, Wnp, out);
}